// GCN_89970974917000
// MI455X (gfx1250) — compile-verified
//
#include <hip/hip_runtime.h>

typedef __attribute__((ext_vector_type(2))) float v2f;
typedef __attribute__((ext_vector_type(8))) float v8f;

#define N_NODES 100000
#define N_EDGES 1600000
#define IN_F    128
#define HID     64
#define N_CLS   4
#define BN_EPS  1e-5f

// ---------------------------------------------------------------- utilities
__global__ void fill_kernel(float* __restrict__ p, int n, float v) {
    int i = blockIdx.x * blockDim.x + threadIdx.x;
    if (i < n) p[i] = v;
}

// degree: deg[dst] += 1 per edge (deg pre-filled with 1.0 for self loops)
__global__ void deg_kernel(const long long* __restrict__ ei, float* __restrict__ deg) {
    int e = blockIdx.x * blockDim.x + threadIdx.x;
    if (e < N_EDGES) {
        int d = (int)ei[(size_t)N_EDGES + e];
        atomicAdd(&deg[d], 1.0f);
    }
}

__global__ void dinv_kernel(const float* __restrict__ deg, float* __restrict__ dinv) {
    int i = blockIdx.x * blockDim.x + threadIdx.x;
    if (i < N_NODES) {
        float d = deg[i];
        dinv[i] = (d > 0.0f) ? rsqrtf(fmaxf(d, 1.0f)) : 0.0f;
    }
}

// ---------------------------------------------------------------- WMMA GEMM
// H[M x outCols] = X[M x K] @ W[K x ldW], W zero-padded to OUT=NT*16 cols in LDS.
// One wave computes a 16 x (NT*16) tile using V_WMMA_F32_16X16X4_F32.
// LDS holds W as k-pairs: ldsW[kp*OUT + col] = {W[2kp][col], W[2kp+1][col]}
// so a B fragment (4x16, K = v + 2*(lane/16)) is a single float2 LDS load.
template <int K, int NT>
__global__ __launch_bounds__(256) void gemm_wmma_kernel(
        const float* __restrict__ X, const float* __restrict__ W,
        float* __restrict__ H, int ldW, int outCols) {
    constexpr int OUT = NT * 16;
    __shared__ v2f ldsW[(K / 2) * OUT];

    for (int idx = threadIdx.x; idx < (K / 2) * OUT; idx += blockDim.x) {
        int kp = idx / OUT, col = idx % OUT;
        v2f w;
        if (col < ldW) { w.x = W[(2 * kp) * ldW + col]; w.y = W[(2 * kp + 1) * ldW + col]; }
        else           { w.x = 0.0f;                    w.y = 0.0f; }
        ldsW[idx] = w;
    }
    __syncthreads();

    int wave = threadIdx.x >> 5;
    int lane = threadIdx.x & 31;
    int tile = blockIdx.x * (blockDim.x >> 5) + wave;
    int m0   = tile * 16;
    if (m0 >= N_NODES) return;

    int g    = lane >> 4;   // lane group: selects K = {0,1} vs {2,3} within a step
    int lmod = lane & 15;

    v8f zero = {};
    v8f acc[NT];
#pragma unroll
    for (int n = 0; n < NT; ++n) acc[n] = zero;

    const float* xrow = X + (size_t)(m0 + lmod) * K;  // A: lanes 0-15 & 16-31 both hold M=0..15

#pragma unroll 4
    for (int k = 0; k < K; k += 4) {
        v2f a = *(const v2f*)(xrow + k + 2 * g);      // A[v]: K = v + 2g (contiguous pair)
#pragma unroll
        for (int n = 0; n < NT; ++n) {
            v2f b = ldsW[(k / 2 + g) * OUT + n * 16 + lmod];  // B[v]: K = v + 2g
            acc[n] = __builtin_amdgcn_wmma_f32_16x16x4_f32(
                false, a, false, b, (short)0, acc[n], false, false);
        }
    }

#pragma unroll
    for (int n = 0; n < NT; ++n) {
        int col = n * 16 + lmod;
        if (col < outCols) {
#pragma unroll
            for (int v = 0; v < 8; ++v) {             // C/D: row = v + 8g
                int row = m0 + v + 8 * g;
                H[(size_t)row * outCols + col] = acc[n][v];
            }
        }
    }
}

// ---------------------------------------------------------------- aggregation
// out[i][f] = h[i][f] * dinv[i]^2 + bias[f]   (self-loop message + bias)
template <int F>
__global__ void agg_init_kernel(const float* __restrict__ h, const float* __restrict__ dinv,
                                const float* __restrict__ bias, float* __restrict__ out) {
    int t = blockIdx.x * blockDim.x + threadIdx.x;
    if (t < N_NODES * F) {
        int i = t / F, f = t % F;
        float dv = dinv[i];
        out[t] = h[t] * dv * dv + bias[f];
    }
}

// out[dst] += h[src] * dinv[src]*dinv[dst]; 4 features per thread (vector gather)
template <int F>
__global__ void agg_edge_kernel(const long long* __restrict__ ei,
                                const float* __restrict__ h, const float* __restrict__ dinv,
                                float* __restrict__ out) {
    constexpr int F4 = F / 4;
    int t = blockIdx.x * blockDim.x + threadIdx.x;
    if (t >= N_EDGES * F4) return;
    int e = t / F4;
    int q = (t % F4) * 4;
    int s = (int)ei[e];
    int d = (int)ei[(size_t)N_EDGES + e];
    float c = dinv[s] * dinv[d];
    float4 m = *(const float4*)(h + (size_t)s * F + q);
    float* op = out + (size_t)d * F + q;
    atomicAdd(op + 0, m.x * c);
    atomicAdd(op + 1, m.y * c);
    atomicAdd(op + 2, m.z * c);
    atomicAdd(op + 3, m.w * c);
}

// ---------------------------------------------------------------- batch norm
__global__ __launch_bounds__(256) void bn_stats_kernel(const float* __restrict__ x,
                                                       float* __restrict__ sums,
                                                       float* __restrict__ sumsq) {
    int f  = threadIdx.x & 63;
    int ry = threadIdx.x >> 6;           // 0..3
    float s = 0.0f, ss = 0.0f;
    int rowBase = blockIdx.x * 128;
    for (int r = ry; r < 128; r += 4) {
        int row = rowBase + r;
        if (row < N_NODES) {
            float v = x[(size_t)row * 64 + f];
            s += v; ss += v * v;
        }
    }
    __shared__ float sh[2][4][64];
    sh[0][ry][f] = s; sh[1][ry][f] = ss;
    __syncthreads();
    if (ry == 0) {
        s  = sh[0][0][f] + sh[0][1][f] + sh[0][2][f] + sh[0][3][f];
        ss = sh[1][0][f] + sh[1][1][f] + sh[1][2][f] + sh[1][3][f];
        atomicAdd(&sums[f], s);
        atomicAdd(&sumsq[f], ss);
    }
}

__global__ void bn_apply_relu_kernel(float* __restrict__ x,
                                     const float* __restrict__ sums,
                                     const float* __restrict__ sumsq,
                                     const float* __restrict__ gamma,
                                     const float* __restrict__ beta) {
    int t = blockIdx.x * blockDim.x + threadIdx.x;
    if (t < N_NODES * 64) {
        int f = t & 63;
        const float inv_n = 1.0f / (float)N_NODES;
        float mean = sums[f] * inv_n;
        float var  = sumsq[f] * inv_n - mean * mean;   // biased, matches jnp.var
        float r    = rsqrtf(var + BN_EPS);
        float v    = (x[t] - mean) * r * gamma[f] + beta[f];
        x[t] = v > 0.0f ? v : 0.0f;
    }
}

// ---------------------------------------------------------------- launcher
static inline int cdiv(int a, int b) { return (a + b - 1) / b; }

extern "C" void kernel_launch(void* const* d_in, const int* in_sizes, int n_in,
                              void* d_out, int out_size, void* d_ws, size_t ws_size,
                              hipStream_t stream) {
    const float*     x   = (const float*)d_in[0];
    const long long* ei  = (const long long*)d_in[1];
    const float* W1 = (const float*)d_in[2];  const float* b1  = (const float*)d_in[3];
    const float* g1 = (const float*)d_in[4];  const float* be1 = (const float*)d_in[5];
    const float* W2 = (const float*)d_in[6];  const float* b2  = (const float*)d_in[7];
    const float* g2 = (const float*)d_in[8];  const float* be2 = (const float*)d_in[9];
    const float* W3 = (const float*)d_in[10]; const float* b3  = (const float*)d_in[11];
    float* out = (float*)d_out;

    // workspace carve-up (floats); total ~53.6 MB
    float* ws    = (float*)d_ws;
    float* deg   = ws;                         // 100000
    float* dinv  = ws + 100000;                // 100000
    float* sums  = ws + 200000;                // 64
    float* sumsq = ws + 200064;                // 64
    float* hA    = ws + 200128;                // 100000*64
    float* hB    = hA + (size_t)N_NODES * 64;  // 100000*64
    float* h3    = hB + (size_t)N_NODES * 64;  // 100000*4

    const int T = 256;
    const int gemmBlocks = cdiv(N_NODES / 16, 8);         // 6250 tiles / 8 waves = 782

    // --- normalization coefficients ---
    fill_kernel<<<cdiv(N_NODES, T), T, 0, stream>>>(deg, N_NODES, 1.0f); // self loop
    deg_kernel<<<cdiv(N_EDGES, T), T, 0, stream>>>(ei, deg);
    dinv_kernel<<<cdiv(N_NODES, T), T, 0, stream>>>(deg, dinv);

    // --- layer 1: x@W1 -> aggregate -> BN+ReLU ---
    gemm_wmma_kernel<IN_F, 4><<<gemmBlocks, T, 0, stream>>>(x, W1, hA, HID, HID);
    agg_init_kernel<64><<<cdiv(N_NODES * 64, T), T, 0, stream>>>(hA, dinv, b1, hB);
    agg_edge_kernel<64><<<cdiv(N_EDGES * 16, T), T, 0, stream>>>(ei, hA, dinv, hB);
    fill_kernel<<<1, 128, 0, stream>>>(sums, 128, 0.0f);  // sums+sumsq contiguous
    bn_stats_kernel<<<cdiv(N_NODES, 128), T, 0, stream>>>(hB, sums, sumsq);
    bn_apply_relu_kernel<<<cdiv(N_NODES * 64, T), T, 0, stream>>>(hB, sums, sumsq, g1, be1);

    // --- layer 2: h@W2 -> aggregate -> BN+ReLU ---
    gemm_wmma_kernel<HID, 4><<<gemmBlocks, T, 0, stream>>>(hB, W2, hA, HID, HID);
    agg_init_kernel<64><<<cdiv(N_NODES * 64, T), T, 0, stream>>>(hA, dinv, b2, hB);
    agg_edge_kernel<64><<<cdiv(N_EDGES * 16, T), T, 0, stream>>>(ei, hA, dinv, hB);
    fill_kernel<<<1, 128, 0, stream>>>(sums, 128, 0.0f);
    bn_stats_kernel<<<cdiv(N_NODES, 128), T, 0, stream>>>(hB, sums, sumsq);
    bn_apply_relu_kernel<<<cdiv(N_NODES * 64, T), T, 0, stream>>>(hB, sums, sumsq, g2, be2);

    // --- layer 3: h@W3 (N padded 4->16) -> aggregate into d_out ---
    gemm_wmma_kernel<HID, 1><<<gemmBlocks, T, 0, stream>>>(hB, W3, h3, N_CLS, N_CLS);
    agg_init_kernel<4><<<cdiv(N_NODES * 4, T), T, 0, stream>>>(h3, dinv, b3, out);
    agg_edge_kernel<4><<<cdiv(N_EDGES, T), T, 0, stream>>>(ei, h3, dinv, out);
}